// SpatialBlock_2869038154382
// MI455X (gfx1250) — compile-verified
//
#include <hip/hip_runtime.h>
#include <hip/hip_bf16.h>

// Problem constants (match reference)
#define BQ   16
#define TQ   12
#define BT   (BQ * TQ)          // 192
#define NN   512
#define DD   128                 // = H*DH
#define HH_  4
#define DH_  32
#define EE   4096
#define NEG_SLOPE 0.2f
#define LN_EPS    1e-5f

typedef float v2f __attribute__((ext_vector_type(2)));
typedef float v8f __attribute__((ext_vector_type(8)));

// ---------------------------------------------------------------------------
// Fill kernel (zero-init of out / denom / max-keys; 0.0f bit pattern == 0u,
// which is below every encoded float key, i.e. acts as -inf for the max)
// ---------------------------------------------------------------------------
__global__ __launch_bounds__(256) void fill_f32(float* __restrict__ p, float v, int n) {
  int i = blockIdx.x * 256 + threadIdx.x;
  if (i < n) p[i] = v;
}

// ---------------------------------------------------------------------------
// Kernel 1: projection GEMM  h[BT*N, 128] = x[BT*N, 128] @ W[128, 128]
// fp32 WMMA 16x16x4. One wave per 16-row strip, 8 N-tiles per wave,
// K-loop of 32 -> 256 v_wmma_f32_16x16x4_f32 per wave.
// ---------------------------------------------------------------------------
__global__ __launch_bounds__(256)
void gemm_proj_kernel(const float* __restrict__ x, const float* __restrict__ W,
                      float* __restrict__ h, int Mtotal) {
  const int lane = threadIdx.x & 31;
  const int wave = threadIdx.x >> 5;
  const int strip = blockIdx.x * 8 + wave;
  const int row0  = strip * 16;
  if (row0 >= Mtotal) return;   // uniform per-wave: EXEC stays all-ones for WMMA

  const int l15 = lane & 15;
  const int hi  = lane >> 4;    // 0: K pair {k,k+1}; 1: K pair {k+2,k+3}

  v8f acc[8] = {};

  // A: lanes 0-15 hold rows M=0..15 (K=k..k+1), lanes 16-31 same rows (K=k+2..k+3)
  const float* arow = x + (size_t)(row0 + l15) * DD + hi * 2;
  // B: lanes 0-15 hold row K=k   (cols), lanes 16-31 row K=k+2; VGPR1 -> K+1
  const float* bcol = W + (size_t)(hi * 2) * DD + l15;

#pragma unroll
  for (int kt = 0; kt < 32; ++kt) {
    const int k = kt * 4;
    v2f a = *reinterpret_cast<const v2f*>(arow + k);   // 8B-aligned
    const float* bp = bcol + (size_t)k * DD;
#pragma unroll
    for (int nt = 0; nt < 8; ++nt) {
      v2f b;
      b.x = bp[nt * 16];        // K = k + 2*hi
      b.y = bp[nt * 16 + DD];   // K = k + 2*hi + 1
      acc[nt] = __builtin_amdgcn_wmma_f32_16x16x4_f32(
          /*neg_a=*/false, a, /*neg_b=*/false, b,
          /*c_mod=*/(short)0, acc[nt], /*reuse_a=*/false, /*reuse_b=*/false);
    }
  }

  // D layout: VGPR v: lanes 0-15 -> (M=v, N=lane), lanes 16-31 -> (M=v+8, N=lane-16)
  float* hrow = h + (size_t)(row0 + hi * 8) * DD + l15;
#pragma unroll
  for (int v = 0; v < 8; ++v) {
#pragma unroll
    for (int nt = 0; nt < 8; ++nt) {
      hrow[(size_t)v * DD + nt * 16] = acc[nt][v];
    }
  }
}

// ---------------------------------------------------------------------------
// Kernel 2: per-node attention scores  s_src/s_dst[BT*N, H]
// ---------------------------------------------------------------------------
__global__ __launch_bounds__(256)
void scores_kernel(const float* __restrict__ h, const float* __restrict__ att_src,
                   const float* __restrict__ att_dst, float* __restrict__ s_src,
                   float* __restrict__ s_dst, int total) {   // total = BT*N*H
  int idx = blockIdx.x * 256 + threadIdx.x;
  if (idx >= total) return;
  int hh = idx & (HH_ - 1);
  const float* hp = h + (size_t)(idx >> 2) * DD + hh * DH_;
  const float* as = att_src + hh * DH_;
  const float* ad = att_dst + hh * DH_;
  float ss = 0.f, sd = 0.f;
#pragma unroll
  for (int i = 0; i < DH_; i += 4) {
    float4 v = *reinterpret_cast<const float4*>(hp + i);
    float4 a = *reinterpret_cast<const float4*>(as + i);
    float4 b = *reinterpret_cast<const float4*>(ad + i);
    ss += v.x * a.x + v.y * a.y + v.z * a.z + v.w * a.w;
    sd += v.x * b.x + v.y * b.y + v.z * b.z + v.w * b.w;
  }
  s_src[idx] = ss;
  s_dst[idx] = sd;
}

// ordered-float <-> uint key for atomic max over possibly-negative floats
__device__ __forceinline__ unsigned int f2key(float f) {
  unsigned int b = __float_as_uint(f);
  return (b & 0x80000000u) ? ~b : (b | 0x80000000u);
}
__device__ __forceinline__ float key2f(unsigned int k) {
  unsigned int b = (k & 0x80000000u) ? (k ^ 0x80000000u) : ~k;
  return __uint_as_float(b);
}

// ---------------------------------------------------------------------------
// Kernel 3: edge logits (leaky_relu, * edge_weight) + scatter-max per dst
// ---------------------------------------------------------------------------
__global__ __launch_bounds__(256)
void edge_logits_kernel(const float* __restrict__ s_src, const float* __restrict__ s_dst,
                        const int* __restrict__ edge_index,
                        const float* __restrict__ edge_weight,
                        float* __restrict__ logits, unsigned int* __restrict__ mkey,
                        int total) {   // total = BT*E*H
  int idx = blockIdx.x * 256 + threadIdx.x;
  if (idx >= total) return;
  int hh = idx & (HH_ - 1);
  int e  = (idx >> 2) & (EE - 1);
  int bt = idx >> 14;                       // / (E*H)
  int sn = edge_index[e];
  int dn = edge_index[EE + e];
  float v = s_src[((size_t)bt * NN + sn) * HH_ + hh] +
            s_dst[((size_t)bt * NN + dn) * HH_ + hh];
  v = v > 0.f ? v : NEG_SLOPE * v;
  v *= edge_weight[e];
  logits[idx] = v;
  atomicMax(&mkey[((size_t)bt * NN + dn) * HH_ + hh], f2key(v));
}

// ---------------------------------------------------------------------------
// Kernel 4: ex = exp(logit - max[dst]); scatter-add into denom (ex overwrites logits)
// ---------------------------------------------------------------------------
__global__ __launch_bounds__(256)
void edge_exp_kernel(float* __restrict__ logits, const unsigned int* __restrict__ mkey,
                     const int* __restrict__ edge_index, float* __restrict__ denom,
                     int total) {   // total = BT*E*H
  int idx = blockIdx.x * 256 + threadIdx.x;
  if (idx >= total) return;
  int hh = idx & (HH_ - 1);
  int e  = (idx >> 2) & (EE - 1);
  int bt = idx >> 14;
  int dn = edge_index[EE + e];
  float m  = key2f(mkey[((size_t)bt * NN + dn) * HH_ + hh]);
  float ex = expf(logits[idx] - m);
  logits[idx] = ex;
  atomicAdd(&denom[((size_t)bt * NN + dn) * HH_ + hh], ex);
}

// ---------------------------------------------------------------------------
// Kernel 5: alpha = ex / (denom[dst]+1e-16); out[dst, :] += alpha * h[src, :]
// 2 edges per 256-thread block, 128 features per edge.
// ---------------------------------------------------------------------------
__global__ __launch_bounds__(256)
void aggregate_kernel(const float* __restrict__ h, const float* __restrict__ ex,
                      const float* __restrict__ denom, const int* __restrict__ edge_index,
                      float* __restrict__ out, int totalE) {   // totalE = BT*E
  int ee = blockIdx.x * 2 + (threadIdx.x >> 7);
  if (ee >= totalE) return;
  int f  = threadIdx.x & 127;
  int e  = ee & (EE - 1);
  int bt = ee >> 12;
  int sn = edge_index[e];
  int dn = edge_index[EE + e];
  int hh = f >> 5;
  float a = ex[(size_t)ee * HH_ + hh];
  float d = denom[((size_t)bt * NN + dn) * HH_ + hh];
  float alpha = a / (d + 1e-16f);
  float val = alpha * h[((size_t)bt * NN + sn) * DD + f];
  atomicAdd(&out[((size_t)bt * NN + dn) * DD + f], val);
}

// ---------------------------------------------------------------------------
// Kernel 6: +bias, LayerNorm, ELU (in place on out). One wave32 per node,
// 4 features per lane, butterfly shuffle reductions.
// ---------------------------------------------------------------------------
__global__ __launch_bounds__(256)
void ln_elu_kernel(float* __restrict__ out, const float* __restrict__ bias,
                   const float* __restrict__ gamma, const float* __restrict__ beta,
                   int nodes) {   // nodes = BT*N
  int node = blockIdx.x * 8 + (threadIdx.x >> 5);
  if (node >= nodes) return;
  int lane = threadIdx.x & 31;
  float* p = out + (size_t)node * DD + lane * 4;
  float4 v  = *reinterpret_cast<float4*>(p);
  float4 bb = *reinterpret_cast<const float4*>(bias + lane * 4);
  v.x += bb.x; v.y += bb.y; v.z += bb.z; v.w += bb.w;

  float s = v.x + v.y + v.z + v.w;
#pragma unroll
  for (int off = 16; off > 0; off >>= 1) s += __shfl_xor(s, off, 32);
  float mu = s * (1.0f / DD);

  float dx = v.x - mu, dy = v.y - mu, dz = v.z - mu, dw = v.w - mu;
  float q = dx * dx + dy * dy + dz * dz + dw * dw;
#pragma unroll
  for (int off = 16; off > 0; off >>= 1) q += __shfl_xor(q, off, 32);
  float rstd = rsqrtf(q * (1.0f / DD) + LN_EPS);

  float4 g  = *reinterpret_cast<const float4*>(gamma + lane * 4);
  float4 be = *reinterpret_cast<const float4*>(beta + lane * 4);
  float y;
  y = dx * rstd * g.x + be.x; v.x = y > 0.f ? y : expm1f(y);
  y = dy * rstd * g.y + be.y; v.y = y > 0.f ? y : expm1f(y);
  y = dz * rstd * g.z + be.z; v.z = y > 0.f ? y : expm1f(y);
  y = dw * rstd * g.w + be.w; v.w = y > 0.f ? y : expm1f(y);
  *reinterpret_cast<float4*>(p) = v;
}

// ---------------------------------------------------------------------------
extern "C" void kernel_launch(void* const* d_in, const int* in_sizes, int n_in,
                              void* d_out, int out_size, void* d_ws, size_t ws_size,
                              hipStream_t stream) {
  const float* x           = (const float*)d_in[0];
  const int*   edge_index  = (const int*)  d_in[1];
  const float* edge_weight = (const float*)d_in[2];
  const float* W           = (const float*)d_in[3];
  const float* att_src     = (const float*)d_in[4];
  const float* att_dst     = (const float*)d_in[5];
  const float* bias        = (const float*)d_in[6];
  const float* ln_gamma    = (const float*)d_in[7];
  const float* ln_beta     = (const float*)d_in[8];
  float* out = (float*)d_out;

  const int nodes   = BT * NN;           // 98304
  const int hElems  = nodes * DD;        // 12,582,912
  const int sElems  = nodes * HH_;       // 393,216
  const int eElems  = BT * EE * HH_;     // 3,145,728
  const int totalE  = BT * EE;           // 786,432

  // workspace layout (floats)
  float* ws    = (float*)d_ws;
  float* h     = ws;                               // hElems
  float* s_src = h + hElems;                       // sElems
  float* s_dst = s_src + sElems;                   // sElems
  float* mkeyf = s_dst + sElems;                   // sElems (uint keys)
  float* denom = mkeyf + sElems;                   // sElems
  float* exbuf = denom + sElems;                   // eElems (logits then ex)
  unsigned int* mkey = (unsigned int*)mkeyf;

  // init: out = 0 (aggregation target), mkey = 0 (== -inf key), denom = 0
  fill_f32<<<(hElems + 255) / 256, 256, 0, stream>>>(out, 0.0f, hElems);
  fill_f32<<<(2 * sElems + 255) / 256, 256, 0, stream>>>(mkeyf, 0.0f, 2 * sElems);

  // 1) projection GEMM (WMMA fp32)
  gemm_proj_kernel<<<nodes / 16 / 8, 256, 0, stream>>>(x, W, h, nodes);

  // 2) per-node scores
  scores_kernel<<<(sElems + 255) / 256, 256, 0, stream>>>(h, att_src, att_dst,
                                                          s_src, s_dst, sElems);

  // 3) edge logits + scatter max
  edge_logits_kernel<<<(eElems + 255) / 256, 256, 0, stream>>>(
      s_src, s_dst, edge_index, edge_weight, exbuf, mkey, eElems);

  // 4) exp + scatter-add denom
  edge_exp_kernel<<<(eElems + 255) / 256, 256, 0, stream>>>(
      exbuf, mkey, edge_index, denom, eElems);

  // 5) weighted scatter-add aggregation
  aggregate_kernel<<<(totalE + 1) / 2, 256, 0, stream>>>(
      h, exbuf, denom, edge_index, out, totalE);

  // 6) bias + LayerNorm + ELU (in place)
  ln_elu_kernel<<<(nodes + 7) / 8, 256, 0, stream>>>(out, bias, ln_gamma, ln_beta, nodes);
}